// Interpolator_38946763440958
// MI455X (gfx1250) — compile-verified
//
#include <hip/hip_runtime.h>
#include <stdint.h>

// Problem constants (from reference setup_inputs)
#define NH_SZ 9           // neighbor cells per cell
#define L_PTS 16          // fine points per coarse cell
#define NT 16             // target points per cell
#define NV 4              // variables
#define KCAND 144         // NH_SZ * L_PTS candidate source points
#define N_NH 3            // top-k size
#define DIST_STRIDE 148   // padded LDS row stride (floats): 592 B, 16B-aligned rows
#define MASK_VALUE 1.0e6f
#define CUTOFF 0.01f

// One workgroup (64 threads = 2 wave32) per coarse cell.
__global__ __launch_bounds__(64) void interp_topk_kernel(
    const float* __restrict__ x,       // (n, nv)  flat: cell*64 + p*4 + v
    const float* __restrict__ mask,    // (n, nv)  same layout
    const float* __restrict__ dist,    // (n_l, nt, K)
    const int*   __restrict__ nh_idx,  // (n_l, 9)
    float*       __restrict__ out)     // (n_l*nt, nv)
{
    __shared__ __align__(16) float s_dist[NT * DIST_STRIDE];  // 16 x 148
    __shared__ __align__(16) float s_x[KCAND * NV];           // [k][v]
    __shared__ __align__(16) float s_mask[KCAND * NV];        // [k][v] (staging)
    __shared__ __align__(16) float s_maskT[NV * DIST_STRIDE]; // [v][k] padded

    const int i   = blockIdx.x;
    const int tid = threadIdx.x;

    // LDS byte offsets: low 32 bits of the generic pointer are the wave-relative
    // LDS byte address (shared-aperture generic addressing).
    const uint32_t distL = (uint32_t)(uintptr_t)&s_dist[0];
    const uint32_t xL    = (uint32_t)(uintptr_t)&s_x[0];
    const uint32_t mL    = (uint32_t)(uintptr_t)&s_mask[0];

    const uint64_t distG = (uint64_t)(uintptr_t)dist + (uint64_t)i * (NT * KCAND * 4);
    const uint64_t xG    = (uint64_t)(uintptr_t)x;
    const uint64_t mG    = (uint64_t)(uintptr_t)mask;

    // ---- Stage dist tile: 16 rows x 36 float4 (contiguous in memory),
    //      written into padded LDS rows of 37 float4. 576 B128 async loads. ----
    #pragma unroll
    for (int r = 0; r < 9; ++r) {
        const int j   = r * 64 + tid;      // 0..575
        const int row = j / 36;
        const int w   = j - row * 36;
        const uint32_t lds = distL + (uint32_t)(row * 37 + w) * 16u;
        const uint32_t off = (uint32_t)j * 16u;
        asm volatile("global_load_async_to_lds_b128 %0, %1, %2"
                     :: "v"(lds), "v"(off), "s"(distG) : "memory");
    }

    // ---- Stage gathered x chunks: 9 neighbor cells x 16 float4 = 144 B128. ----
    #pragma unroll
    for (int r = 0; r < 3; ++r) {
        const int j = r * 64 + tid;        // 0..191, valid < 144
        if (j < KCAND) {
            const int c  = j >> 4;         // neighbor slot 0..8
            const int w  = j & 15;         // float4 within 64-float chunk
            const int nc = nh_idx[i * NH_SZ + c];
            const uint32_t off = (uint32_t)nc * 256u + (uint32_t)w * 16u;
            const uint32_t lds = xL + (uint32_t)j * 16u;
            asm volatile("global_load_async_to_lds_b128 %0, %1, %2"
                         :: "v"(lds), "v"(off), "s"(xG) : "memory");
        }
    }
    // ---- Stage gathered mask chunks (same addressing, uniform base). ----
    #pragma unroll
    for (int r = 0; r < 3; ++r) {
        const int j = r * 64 + tid;
        if (j < KCAND) {
            const int c  = j >> 4;
            const int w  = j & 15;
            const int nc = nh_idx[i * NH_SZ + c];
            const uint32_t off = (uint32_t)nc * 256u + (uint32_t)w * 16u;
            const uint32_t lds = mL + (uint32_t)j * 16u;
            asm volatile("global_load_async_to_lds_b128 %0, %1, %2"
                         :: "v"(lds), "v"(off), "s"(mG) : "memory");
        }
    }

    asm volatile("s_wait_asynccnt 0x0" ::: "memory");
    __syncthreads();

    // ---- Transpose mask [k][v] -> [v][k] (padded) so the scan can use b128. ----
    #pragma unroll
    for (int r = 0; r < 9; ++r) {
        const int idx = r * 64 + tid;      // 0..575  == k*4 + v
        const int k   = idx >> 2;
        const int v   = idx & 3;
        s_maskT[v * DIST_STRIDE + k] = s_mask[idx];
    }
    __syncthreads();

    // ---- Per-lane top-3 scan over K=144 masked distances. lane = (t,v). ----
    const int t = tid >> 2;
    const int v = tid & 3;
    const float* dRow = &s_dist[t * DIST_STRIDE];
    const float* mRow = &s_maskT[v * DIST_STRIDE];

    float d0 = 3.4e38f, d1 = 3.4e38f, d2 = 3.4e38f;
    int   k0 = 0,       k1 = 0,       k2 = 0;

    auto insert = [&](float dm, int k) {
        // Strict < keeps earlier index on ties (matches jax.lax.top_k).
        const bool c0 = dm < d0;
        const bool c1 = dm < d1;
        const bool c2 = dm < d2;
        const float s2 = c2 ? dm : d2;      // candidate for slot 2 if not promoted
        const int   j2 = c2 ? k  : k2;
        const float nd2 = c1 ? d1 : s2;
        const int   nk2 = c1 ? k1 : j2;
        const float nd1 = c0 ? d0 : (c1 ? dm : d1);
        const int   nk1 = c0 ? k0 : (c1 ? k  : k1);
        d0 = c0 ? dm : d0;  k0 = c0 ? k : k0;
        d1 = nd1;           k1 = nk1;
        d2 = nd2;           k2 = nk2;
    };

    #pragma unroll 2
    for (int kb = 0; kb < KCAND; kb += 4) {
        const float4 dd = *(const float4*)(dRow + kb);   // ds_load_b128
        const float4 mm = *(const float4*)(mRow + kb);   // ds_load_b128
        insert(fmaf(mm.x, MASK_VALUE, dd.x), kb + 0);
        insert(fmaf(mm.y, MASK_VALUE, dd.y), kb + 1);
        insert(fmaf(mm.z, MASK_VALUE, dd.z), kb + 2);
        insert(fmaf(mm.w, MASK_VALUE, dd.w), kb + 3);
    }

    // ---- Inverse-distance-squared weights + gather + normalize. ----
    const float e0 = fmaxf(d0, CUTOFF);
    const float e1 = fmaxf(d1, CUTOFF);
    const float e2 = fmaxf(d2, CUTOFF);
    const float w0 = 1.0f / (e0 * e0);
    const float w1 = 1.0f / (e1 * e1);
    const float w2 = 1.0f / (e2 * e2);
    const float ws = w0 + w1 + w2;

    const float xv = s_x[k0 * 4 + v] * w0
                   + s_x[k1 * 4 + v] * w1
                   + s_x[k2 * 4 + v] * w2;

    out[i * (NT * NV) + tid] = xv / ws;
}

extern "C" void kernel_launch(void* const* d_in, const int* in_sizes, int n_in,
                              void* d_out, int out_size, void* d_ws, size_t ws_size,
                              hipStream_t stream) {
    const float* x      = (const float*)d_in[0];
    const float* mask   = (const float*)d_in[1];
    const float* dist   = (const float*)d_in[2];
    const int*   nh_idx = (const int*)d_in[3];
    float*       out    = (float*)d_out;

    const int n_l = in_sizes[3] / NH_SZ;   // 4096 coarse cells
    interp_topk_kernel<<<n_l, 64, 0, stream>>>(x, mask, dist, nh_idx, out);
}